// GAT_16157666968389
// MI455X (gfx1250) — compile-verified
//
#include <hip/hip_runtime.h>
#include <hip/hip_bf16.h>
#include <math.h>

// ---------------------------------------------------------------------------
// GATv2 x4 + pooling + MLP for MI455X (gfx1250, wave32).
// GEMMs use full-precision f32 WMMA (v_wmma_f32_16x16x4_f32) so numerics match
// the f32 reference; problem is L2-resident (<<192MB) so precision downcast
// buys nothing. All reductions are fixed-order (CSR + shfl trees) =>
// deterministic output; only integer atomics are used (count/scatter), and
// within-bucket order is canonicalized by sorting edge ids.
// ---------------------------------------------------------------------------

typedef __attribute__((ext_vector_type(2))) float v2f;
typedef __attribute__((ext_vector_type(8))) float v8f;

// ---------------------------------------------------------------------------
// Generic fill
// ---------------------------------------------------------------------------
__global__ void k_fill_i32(int* __restrict__ p, int v, int n) {
    int i = blockIdx.x * blockDim.x + threadIdx.x;
    if (i < n) p[i] = v;
}

// ---------------------------------------------------------------------------
// Edge materialization (append self-loops) + per-dst degree count
// edge_index is [2,E] int64 row-major: first E = src, next E = dst.
// ---------------------------------------------------------------------------
__global__ void k_build_edges(const long long* __restrict__ ei, int E, int Etot,
                              int* __restrict__ src, int* __restrict__ dst,
                              int* __restrict__ count) {
    int e = blockIdx.x * blockDim.x + threadIdx.x;
    if (e >= Etot) return;
    int s, d;
    if (e < E) { s = (int)ei[e]; d = (int)ei[E + e]; }
    else       { s = d = e - E; }
    src[e] = s;
    dst[e] = d;
    atomicAdd(&count[d], 1);
}

// ---------------------------------------------------------------------------
// Single-block exclusive scan of count[0..Nn) -> offsets[0..Nn]
// blockDim.x must be 1024.
// ---------------------------------------------------------------------------
__global__ void k_scan_offsets(const int* __restrict__ count,
                               int* __restrict__ offsets, int Nn) {
    __shared__ int buf[1024];
    __shared__ int carry;
    const int tid = threadIdx.x;
    if (tid == 0) carry = 0;
    __syncthreads();
    const int nChunks = (Nn + 1023) / 1024;
    for (int ch = 0; ch < nChunks; ++ch) {
        int i = ch * 1024 + tid;
        int v = (i < Nn) ? count[i] : 0;
        buf[tid] = v;
        __syncthreads();
        int base = carry;
        for (int off = 1; off < 1024; off <<= 1) {
            int t = (tid >= off) ? buf[tid - off] : 0;
            __syncthreads();
            if (tid >= off) buf[tid] += t;
            __syncthreads();
        }
        if (i < Nn) offsets[i] = base + buf[tid] - v;  // exclusive
        __syncthreads();
        if (tid == 0) carry = base + buf[1023];
        __syncthreads();
    }
    if (tid == 0) offsets[Nn] = carry;
}

// ---------------------------------------------------------------------------
// Scatter edges into CSR buckets (bucket order canonicalized afterwards)
// ---------------------------------------------------------------------------
__global__ void k_scatter_csr(const int* __restrict__ dst,
                              const int* __restrict__ offsets,
                              int* __restrict__ cursor,
                              int* __restrict__ csr, int Etot) {
    int e = blockIdx.x * blockDim.x + threadIdx.x;
    if (e >= Etot) return;
    int d = dst[e];
    int pos = offsets[d] + atomicAdd(&cursor[d], 1);
    csr[pos] = e;
}

// Insertion-sort each bucket by edge id => deterministic reduction order.
__global__ void k_sort_buckets(const int* __restrict__ offsets,
                               int* __restrict__ csr, int Nn) {
    int n = blockIdx.x * blockDim.x + threadIdx.x;
    if (n >= Nn) return;
    int b = offsets[n], e = offsets[n + 1];
    for (int i = b + 1; i < e; ++i) {
        int v = csr[i];
        int j = i - 1;
        while (j >= b && csr[j] > v) { csr[j + 1] = csr[j]; --j; }
        csr[j + 1] = v;
    }
}

// ---------------------------------------------------------------------------
// Per-graph node ranges (batch is sorted ascending)
// ---------------------------------------------------------------------------
__global__ void k_find_ranges(const long long* __restrict__ batch, int Nn,
                              int* __restrict__ gstart, int* __restrict__ gend) {
    int n = blockIdx.x * blockDim.x + threadIdx.x;
    if (n >= Nn) return;
    int b = (int)batch[n];
    if (n == 0 || (int)batch[n - 1] != b) gstart[b] = n;
    if (n == Nn - 1 || (int)batch[n + 1] != b) gend[b] = n + 1;
}

// ---------------------------------------------------------------------------
// f32 WMMA GEMM: Y[M,F] = X[M,K] * W[F,K]^T + bias[F]
// One wave per 16x16 tile, K stepped by 4 via V_WMMA_F32_16X16X4_F32.
// A-frag (16x4 f32): lanes 0-15 -> M=lane, K=k..k+1 ; lanes 16-31 -> K=k+2..k+3
// B-frag (4x16 f32): lanes 0-15 -> N=lane, K=k..k+1 ; lanes 16-31 -> K=k+2..k+3
//   (B[k][n] = W[n][k], so each lane reads a contiguous pair from a W row)
// C/D (16x16 f32): VGPR r -> M=r (lanes 0-15) / M=r+8 (lanes 16-31), N=lane&15
// Requires M%16==0, F%16==0, K%4==0 (true for all layers here).
// ---------------------------------------------------------------------------
__global__ void k_gemm_xwt_wmma(const float* __restrict__ X,
                                const float* __restrict__ W,
                                const float* __restrict__ bias,
                                float* __restrict__ Y,
                                int M, int K, int F) {
    int wave = (blockIdx.x * blockDim.x + threadIdx.x) >> 5;
    int lane = threadIdx.x & 31;
    int tilesF = F >> 4;
    int tiles  = (M >> 4) * tilesF;
    if (wave >= tiles) return;
    int mTile = wave / tilesF;
    int fTile = wave - mTile * tilesF;
    int row0 = mTile << 4, col0 = fTile << 4;
    int half = lane >> 4;     // 0: K pair (k,k+1)  1: K pair (k+2,k+3)
    int l16  = lane & 15;

    const float* xrow = X + (size_t)(row0 + l16) * K + half * 2;
    const float* wrow = W + (size_t)(col0 + l16) * K + half * 2;

    v8f acc = {};
    for (int k = 0; k < K; k += 4) {
        v2f a = *(const v2f*)(xrow + k);
        v2f b = *(const v2f*)(wrow + k);
        acc = __builtin_amdgcn_wmma_f32_16x16x4_f32(
            /*neg_a=*/false, a, /*neg_b=*/false, b,
            /*c_mod=*/(short)0, acc, /*reuse_a=*/false, /*reuse_b=*/false);
    }

    float bcol = bias[col0 + l16];
#pragma unroll
    for (int r = 0; r < 8; ++r) {
        int row = row0 + r + half * 8;
        Y[(size_t)row * F + col0 + l16] = acc[r] + bcol;
    }
}

// ---------------------------------------------------------------------------
// Edge attention logits: logits[e,h] = sum_c att[h,c]*leaky_relu(xl[src,h,c]+xr[dst,h,c],0.2)
// ---------------------------------------------------------------------------
__global__ void k_edge_logits(const float* __restrict__ xl,
                              const float* __restrict__ xr,
                              const int* __restrict__ src,
                              const int* __restrict__ dst,
                              const float* __restrict__ att,
                              float* __restrict__ logits,
                              int Etot, int H, int C) {
    int t = blockIdx.x * blockDim.x + threadIdx.x;
    if (t >= Etot * H) return;
    int e = t / H, h = t - e * H;
    int F = H * C;
    const float* pl = xl + (size_t)src[e] * F + h * C;
    const float* pr = xr + (size_t)dst[e] * F + h * C;
    const float* pa = att + h * C;
    float acc = 0.f;
    for (int c = 0; c < C; ++c) {
        float s = pl[c] + pr[c];
        float lr = (s > 0.f) ? s : 0.2f * s;
        acc += pa[c] * lr;
    }
    logits[(size_t)e * H + h] = acc;
}

// ---------------------------------------------------------------------------
// Segment softmax + aggregation. One wave per (node, head), lane = channel (C==32).
// Fixed-order shfl trees + sequential bucket walk => deterministic.
// act: 0 = identity, 1 = ELU (alpha=1)
// ---------------------------------------------------------------------------
__global__ void k_gat_aggregate(const float* __restrict__ xl,
                                const float* __restrict__ logits,
                                const int* __restrict__ src,
                                const int* __restrict__ csr,
                                const int* __restrict__ offsets,
                                const float* __restrict__ bias,
                                float* __restrict__ out,
                                int Nn, int H, int C, int act) {
    int gw   = (blockIdx.x * blockDim.x + threadIdx.x) >> 5;
    int lane = threadIdx.x & 31;
    if (gw >= Nn * H) return;
    int n = gw / H, h = gw - n * H;
    int beg = offsets[n], end = offsets[n + 1];
    int F = H * C;

    // pass 1: segment max (fixed lane partition + fixed reduce tree)
    float mx = -3.402823466e38f;
    for (int i = beg + lane; i < end; i += 32)
        mx = fmaxf(mx, logits[(size_t)csr[i] * H + h]);
#pragma unroll
    for (int o = 16; o; o >>= 1) mx = fmaxf(mx, __shfl_xor(mx, o, 32));

    // pass 2: softmax denominator
    float sm = 0.f;
    for (int i = beg + lane; i < end; i += 32)
        sm += expf(logits[(size_t)csr[i] * H + h] - mx);
#pragma unroll
    for (int o = 16; o; o >>= 1) sm += __shfl_xor(sm, o, 32);
    float inv = 1.f / (sm + 1e-16f);

    // pass 3: weighted gather (sequential over sorted bucket; lane = channel)
    float acc = 0.f;
    for (int i = beg; i < end; ++i) {
        int e = csr[i];
        float w = expf(logits[(size_t)e * H + h] - mx) * inv;
        acc = fmaf(w, xl[(size_t)src[e] * F + h * C + lane], acc);
    }
    acc += bias[h * C + lane];
    if (act) acc = (acc > 0.f) ? acc : (expf(acc) - 1.f);  // ELU
    out[(size_t)n * F + h * C + lane] = acc;
}

// ---------------------------------------------------------------------------
// Graph pooling + concat: pooled[g, 0:256]=sum h1, [256:288]=h2, [288:320]=h3, [320:352]=h4
// One block per graph, thread j = output column; sequential node loop => deterministic.
// ---------------------------------------------------------------------------
__global__ void k_pool_concat(const float* __restrict__ h1,
                              const float* __restrict__ h2,
                              const float* __restrict__ h3,
                              const float* __restrict__ h4,
                              const int* __restrict__ gstart,
                              const int* __restrict__ gend,
                              float* __restrict__ pooled) {
    int g = blockIdx.x;
    int j = threadIdx.x;
    if (j >= 352) return;
    const float* srcp;
    int col, stride;
    if (j < 256)      { srcp = h1; col = j;       stride = 256; }
    else if (j < 288) { srcp = h2; col = j - 256; stride = 32;  }
    else if (j < 320) { srcp = h3; col = j - 288; stride = 32;  }
    else              { srcp = h4; col = j - 320; stride = 32;  }
    float acc = 0.f;
    int b = gstart[g], e = gend[g];
    for (int n = b; n < e; ++n) acc += srcp[(size_t)n * stride + col];
    pooled[g * 352 + j] = acc;
}

// ---------------------------------------------------------------------------
// Small dense layer: out[g,o] = act( in[g,:K] . W[o,:K] + b[o] )
// ---------------------------------------------------------------------------
__global__ void k_linear(const float* __restrict__ in,
                         const float* __restrict__ W,
                         const float* __restrict__ b,
                         float* __restrict__ out,
                         int G, int K, int O, int relu) {
    int t = blockIdx.x * blockDim.x + threadIdx.x;
    if (t >= G * O) return;
    int g = t / O, o = t - g * O;
    const float* pi = in + (size_t)g * K;
    const float* pw = W + (size_t)o * K;
    float acc = b[o];
    for (int k = 0; k < K; ++k) acc = fmaf(pi[k], pw[k], acc);
    if (relu && acc < 0.f) acc = 0.f;
    out[(size_t)g * O + o] = acc;
}

// ---------------------------------------------------------------------------
// Write outputs: d_out[0 : G*O) = h ; d_out[G*O : 2*G*O) = log_softmax(h)
// ---------------------------------------------------------------------------
__global__ void k_finalize(const float* __restrict__ h,
                           float* __restrict__ out, int G, int O) {
    int g = blockIdx.x * blockDim.x + threadIdx.x;
    if (g >= G) return;
    const float* row = h + (size_t)g * O;
    float mx = -3.402823466e38f;
    for (int o = 0; o < O; ++o) mx = fmaxf(mx, row[o]);
    float s = 0.f;
    for (int o = 0; o < O; ++o) s += expf(row[o] - mx);
    float lse = mx + logf(s);
    for (int o = 0; o < O; ++o) {
        out[(size_t)g * O + o] = row[o];
        out[(size_t)G * O + (size_t)g * O + o] = row[o] - lse;
    }
}

// ---------------------------------------------------------------------------
// Host side
// ---------------------------------------------------------------------------
static inline dim3 gridFor(long long threads, int block) {
    return dim3((unsigned)((threads + block - 1) / block));
}

extern "C" void kernel_launch(void* const* d_in, const int* in_sizes, int n_in,
                              void* d_out, int out_size, void* d_ws, size_t ws_size,
                              hipStream_t stream) {
    (void)n_in; (void)out_size; (void)ws_size;

    // --- inputs (setup_inputs() dict order, tuples flattened) ---
    const float*     x      = (const float*)d_in[0];
    const float*     g1Wl   = (const float*)d_in[1];
    const float*     g1bl   = (const float*)d_in[2];
    const float*     g1Wr   = (const float*)d_in[3];
    const float*     g1br   = (const float*)d_in[4];
    const float*     g1att  = (const float*)d_in[5];
    const float*     g1bias = (const float*)d_in[6];
    const float*     g2Wl   = (const float*)d_in[7];
    const float*     g2bl   = (const float*)d_in[8];
    const float*     g2Wr   = (const float*)d_in[9];
    const float*     g2br   = (const float*)d_in[10];
    const float*     g2att  = (const float*)d_in[11];
    const float*     g2bias = (const float*)d_in[12];
    const float*     g3Wl   = (const float*)d_in[13];
    const float*     g3bl   = (const float*)d_in[14];
    const float*     g3Wr   = (const float*)d_in[15];
    const float*     g3br   = (const float*)d_in[16];
    const float*     g3att  = (const float*)d_in[17];
    const float*     g3bias = (const float*)d_in[18];
    const float*     lin1w  = (const float*)d_in[19];
    const float*     lin1b  = (const float*)d_in[20];
    const float*     lin2w  = (const float*)d_in[21];
    const float*     lin2b  = (const float*)d_in[22];
    const long long* ei     = (const long long*)d_in[23];
    const long long* batch  = (const long long*)d_in[24];

    const int DIM_IN = 128, C = 32, H1 = 8, F1 = H1 * C /*256*/, F2 = 32;
    const int G = 64, OUTF = 352, OCLS = 16;
    const int N    = in_sizes[0] / DIM_IN;   // 20000
    const int E    = in_sizes[23] / 2;       // 320000
    const int Etot = E + N;                  // self-loops appended

    // --- workspace carving (256B aligned) ---
    char* w = (char*)d_ws;
    size_t off = 0;
    auto alloc = [&](size_t bytes) -> void* {
        void* p = w + off;
        off += (bytes + 255) & ~(size_t)255;
        return p;
    };
    float* xl      = (float*)alloc((size_t)N * F1 * 4);      // reused for all layers
    float* xr      = (float*)alloc((size_t)N * F1 * 4);
    float* h1      = (float*)alloc((size_t)N * F1 * 4);
    float* h2      = (float*)alloc((size_t)N * F2 * 4);
    float* h3      = (float*)alloc((size_t)N * F2 * 4);
    float* h4      = (float*)alloc((size_t)N * F2 * 4);
    float* logits  = (float*)alloc((size_t)Etot * H1 * 4);   // reused (H=1 later)
    int*   srcA    = (int*)alloc((size_t)Etot * 4);
    int*   dstA    = (int*)alloc((size_t)Etot * 4);
    int*   csr     = (int*)alloc((size_t)Etot * 4);
    int*   count   = (int*)alloc((size_t)N * 4);
    int*   cursor  = (int*)alloc((size_t)N * 4);
    int*   offs    = (int*)alloc((size_t)(N + 1) * 4);
    int*   gstart  = (int*)alloc((size_t)G * 4);
    int*   gend    = (int*)alloc((size_t)G * 4);
    float* pooled  = (float*)alloc((size_t)G * OUTF * 4);
    float* hmid    = (float*)alloc((size_t)G * OUTF * 4);
    float* hout    = (float*)alloc((size_t)G * OCLS * 4);

    const int B = 256;

    // --- CSR build (once; shared by all 4 conv layers) ---
    k_fill_i32<<<gridFor(N, B), B, 0, stream>>>(count, 0, N);
    k_fill_i32<<<gridFor(N, B), B, 0, stream>>>(cursor, 0, N);
    k_fill_i32<<<gridFor(G, B), B, 0, stream>>>(gstart, 0, G);
    k_fill_i32<<<gridFor(G, B), B, 0, stream>>>(gend, 0, G);
    k_build_edges<<<gridFor(Etot, B), B, 0, stream>>>(ei, E, Etot, srcA, dstA, count);
    k_scan_offsets<<<1, 1024, 0, stream>>>(count, offs, N);
    k_scatter_csr<<<gridFor(Etot, B), B, 0, stream>>>(dstA, offs, cursor, csr, Etot);
    k_sort_buckets<<<gridFor(N, B), B, 0, stream>>>(offs, csr, N);
    k_find_ranges<<<gridFor(N, B), B, 0, stream>>>(batch, N, gstart, gend);

    auto gemm = [&](const float* X, const float* W_, const float* bias_,
                    float* Y, int M, int K, int F) {
        long long waves = (long long)(M >> 4) * (F >> 4);
        k_gemm_xwt_wmma<<<gridFor(waves * 32, B), B, 0, stream>>>(X, W_, bias_, Y, M, K, F);
    };
    auto conv = [&](const float* Xin, int K, int H, int F,
                    const float* Wl_, const float* bl_,
                    const float* Wr_, const float* br_,
                    const float* att_, const float* bias_,
                    float* Hout, int act) {
        gemm(Xin, Wl_, bl_, xl, N, K, F);
        gemm(Xin, Wr_, br_, xr, N, K, F);
        k_edge_logits<<<gridFor((long long)Etot * H, B), B, 0, stream>>>(
            xl, xr, srcA, dstA, att_, logits, Etot, H, C);
        k_gat_aggregate<<<gridFor((long long)N * H * 32, B), B, 0, stream>>>(
            xl, logits, srcA, csr, offs, bias_, Hout, N, H, C, act);
    };

    // --- 4 GATv2 layers (layer 4 reuses g3 params, matching the reference) ---
    conv(x,  DIM_IN, H1, F1, g1Wl, g1bl, g1Wr, g1br, g1att, g1bias, h1, /*elu*/1);
    conv(h1, F1,     1,  F2, g2Wl, g2bl, g2Wr, g2br, g2att, g2bias, h2, 0);
    conv(h2, F2,     1,  F2, g3Wl, g3bl, g3Wr, g3br, g3att, g3bias, h3, 0);
    conv(h3, F2,     1,  F2, g3Wl, g3bl, g3Wr, g3br, g3att, g3bias, h4, 0);

    // --- pooling + MLP head ---
    k_pool_concat<<<dim3(G), dim3(384), 0, stream>>>(h1, h2, h3, h4, gstart, gend, pooled);
    k_linear<<<gridFor((long long)G * OUTF, B), B, 0, stream>>>(
        pooled, lin1w, lin1b, hmid, G, OUTF, OUTF, /*relu*/1);
    k_linear<<<gridFor((long long)G * OCLS, B), B, 0, stream>>>(
        hmid, lin2w, lin2b, hout, G, OUTF, OCLS, 0);
    k_finalize<<<gridFor(G, 64), 64, 0, stream>>>(hout, (float*)d_out, G, OCLS);
}